// ImagePaste_77841987273276
// MI455X (gfx1250) — compile-verified
//
#include <hip/hip_runtime.h>

// ImagePaste: out[b, r, c, ch] = inside_box ? color[b][ch] : 255.0f
// B = 8192 images, S = 72, 3 channels. Output = 509.6 MB float32 -> pure
// store-bandwidth bound (~22 us at 23.3 TB/s). Strategy: per image, build the
// two possible 216-float row patterns in LDS once, then replicate them to
// global memory with coalesced 128-bit stores via the CDNA5 async LDS->global
// DMA path (GLOBAL_STORE_ASYNC_FROM_LDS_B128, tracked by ASYNCcnt).

#define S_DIM      72
#define ROW_FLOATS (S_DIM * 3)          // 216 floats per image row
#define ROW_VEC4   (ROW_FLOATS / 4)     // 54 float4 per row
#define IMG_FLOATS (S_DIM * S_DIM * 3)  // 15552 floats per image
#define IMG_VEC4   (IMG_FLOATS / 4)     // 3888 float4 per image
#define T_VAL      255.0f

#if defined(__gfx1250__) && __has_builtin(__builtin_amdgcn_global_store_async_from_lds_b128)
#define USE_ASYNC_LDS_STORE 1
#else
#define USE_ASYNC_LDS_STORE 0
#endif

typedef int v4i __attribute__((ext_vector_type(4)));
typedef __attribute__((address_space(1))) v4i as1_v4i;
typedef __attribute__((address_space(3))) v4i as3_v4i;

__global__ __launch_bounds__(256) void ImagePaste_77841987273276_kernel(
    const int*   __restrict__ positions,  // (B,4): tl_r, tl_c, br_r, br_c
    const float* __restrict__ color,      // (B,3)
    float*       __restrict__ out)        // (B,72,72,3)
{
  const int b   = blockIdx.x;
  const int tid = threadIdx.x;

  // Uniform per-block loads -> scalar loads.
  const int p0 = positions[4 * b + 0];  // tl_r
  const int p1 = positions[4 * b + 1];  // tl_c
  const int p2 = positions[4 * b + 2];  // br_r
  const int p3 = positions[4 * b + 3];  // br_c

  // pat[0 .. 215]   : "inside" row (color where col in [tl_c, br_c), else 255)
  // pat[216 .. 431] : "outside" row (all 255)
  __shared__ __align__(16) float pat[2 * ROW_FLOATS];

  if (tid < ROW_FLOATS) {
    const int c  = tid / 3;        // column 0..71
    const int ch = tid - 3 * c;    // channel 0..2
    const bool in_c = (c >= p1) && (c < p3);
    const float cv = color[3 * b + ch];
    pat[tid]              = in_c ? cv : T_VAL;
    pat[ROW_FLOATS + tid] = T_VAL;
  }
  __syncthreads();

  float* dst = out + (size_t)b * IMG_FLOATS;

  // 3888 float4 slots per image; consecutive threads -> consecutive 16B,
  // 512 B contiguous per wave32 per store instruction.
  #pragma unroll 4
  for (int i = tid; i < IMG_VEC4; i += 256) {
    const int r = i / ROW_VEC4;          // row 0..71 (magic-mul division)
    const int j = i - r * ROW_VEC4;      // float4 index within row
    const bool row_in = (r >= p0) && (r < p2);
    const int lds_off = (row_in ? 0 : ROW_FLOATS) + 4 * j;  // float index in pat[]
#if USE_ASYNC_LDS_STORE
    // CDNA5 async DMA: LDS -> global, 16 B per lane, tracked by ASYNCcnt.
    // Two-step casts: pointee reinterpret, then pure addrspace cast.
    __builtin_amdgcn_global_store_async_from_lds_b128(
        (as1_v4i*)(v4i*)(dst + 4 * i),
        (as3_v4i*)(v4i*)(pat + lds_off),
        /*offset=*/0, /*cpol=*/0);
#else
    *(float4*)(dst + 4 * i) = *(const float4*)(pat + lds_off);
#endif
  }

#if USE_ASYNC_LDS_STORE
#if __has_builtin(__builtin_amdgcn_s_wait_asynccnt)
  __builtin_amdgcn_s_wait_asynccnt(0);
#endif
#endif
}

extern "C" void kernel_launch(void* const* d_in, const int* in_sizes, int n_in,
                              void* d_out, int out_size, void* d_ws, size_t ws_size,
                              hipStream_t stream) {
  (void)n_in; (void)out_size; (void)d_ws; (void)ws_size;
  const int*   positions = (const int*)d_in[0];    // (B,4) int32
  const float* color     = (const float*)d_in[1];  // (B,3) float32
  float*       out       = (float*)d_out;          // (B,72,72,3) float32

  const int B = in_sizes[0] / 4;
  ImagePaste_77841987273276_kernel<<<dim3(B), dim3(256), 0, stream>>>(
      positions, color, out);
}